// YoloLayer_42674795053767
// MI455X (gfx1250) — compile-verified
//
#include <hip/hip_runtime.h>
#include <hip/hip_bf16.h>
#include <math.h>

// ---------------- constants ----------------
#define NC       80
#define NA       3
#define HH       76
#define WW       76
#define HW       (HH * WW)          // 5776
#define PP       (NA * HW)          // 17328
#define NT       20
#define STRIDE   8.0f
#define INV_S    0.125f
#define BCE_CAP  27.6310211159f     // -log(1e-12)

__device__ __constant__ float c_aw[9] = {10.f, 16.f, 33.f, 30.f, 62.f, 59.f, 116.f, 156.f, 373.f};
__device__ __constant__ float c_ah[9] = {13.f, 30.f, 23.f, 61.f, 45.f, 119.f, 90.f, 198.f, 326.f};

typedef __attribute__((ext_vector_type(2))) float v2f;
typedef __attribute__((ext_vector_type(8))) float v8f;

// ---------------- helpers ----------------
__device__ __forceinline__ float sigmoidf(float z) { return 1.0f / (1.0f + expf(-z)); }

// stable softplus log(1+exp(z)), capped to match reference's clip(p,1e-12,1-1e-12)
__device__ __forceinline__ float softplus_capped(float z) {
    float sp = fmaxf(z, 0.0f) + log1pf(expf(-fabsf(z)));
    return fminf(sp, BCE_CAP);
}

// Exact f32 wave32 sum via V_WMMA_F32_16X16X4_F32 with a ones B-matrix.
// A layout (16x4 f32): lanes 0-15 hold A[m][0] in VGPR0, A[m][1] in VGPR1;
// lanes 16-31 hold A[m][2], A[m][3]. With a=(v,0): rowsum_m = v_m + v_{m+16}.
// D[m][n] = rowsum_m for all n. Each lane sums its 8 D VGPRs -> half-wave sum;
// shfl_xor(16) adds the other half. EXEC must be all 1s at the call site.
__device__ __forceinline__ float wave_reduce_wmma(float v) {
    v2f a; a[0] = v;    a[1] = 0.0f;
    v2f b; b[0] = 1.0f; b[1] = 1.0f;
    v8f c = {};
    c = __builtin_amdgcn_wmma_f32_16x16x4_f32(false, a, false, b,
                                              (short)0, c, false, false);
    float s = ((c[0] + c[1]) + (c[2] + c[3])) + ((c[4] + c[5]) + (c[6] + c[7]));
    s += __shfl_xor(s, 16, 32);
    return s;
}

// ---------------- kernel 0: zero the 3 output scalars ----------------
__global__ void zero_out_kernel(float* out, int n) {
    int i = blockIdx.x * blockDim.x + threadIdx.x;
    if (i < n) out[i] = 0.0f;
}

// ---------------- kernel 1: per-target matching table ----------------
// ws layout: fdata[B][NT][11] floats, then flat[B][NT] ints.
// fdata fields: 0 valid, 1..4 gx,gy,gw,gh (px), 5 tx, 6 ty, 7 tw, 8 th, 9 cls, 10 scale
__global__ void match_kernel(const float* __restrict__ tgt,
                             float* __restrict__ fdata,
                             int*   __restrict__ flatArr) {
    int b = blockIdx.x;
    int t = threadIdx.x;
    if (t >= NT) return;

    const float* tg = tgt + (size_t)(b * NT + t) * 5;
    float cxn = tg[0], cyn = tg[1], wn = tg[2], hn = tg[3], cls = tg[4];
    bool valid = wn > 0.0f;
    float gx = cxn * 608.0f, gy = cyn * 608.0f;
    float gw = wn * 608.0f,  gh = hn * 608.0f;

    // anchor argmax (first max wins, like jnp.argmax)
    float best = -1.0f; int bi = 0;
    float garea = gw * gh;
    #pragma unroll
    for (int k = 0; k < 9; ++k) {
        float inter = fminf(gw, c_aw[k]) * fminf(gh, c_ah[k]);
        float aiou  = inter / (garea + c_aw[k] * c_ah[k] - inter + 1e-9f);
        if (aiou > best) { best = aiou; bi = k; }
    }
    bool ok = valid && (bi <= 2);

    float gxs = gx * INV_S, gys = gy * INV_S;
    int gi = min(max((int)floorf(gxs), 0), WW - 1);
    int gj = min(max((int)floorf(gys), 0), HH - 1);
    int flat = ok ? (bi * HW + gj * WW + gi) : -1;

    float sw = ok ? gw : c_aw[bi];   // avoid log(0) for invalid rows (unused anyway)
    float sh = ok ? gh : c_ah[bi];

    float* fd = fdata + (size_t)(b * NT + t) * 11;
    fd[0]  = valid ? 1.0f : 0.0f;
    fd[1]  = gx;  fd[2] = gy;  fd[3] = gw;  fd[4] = gh;
    fd[5]  = gxs - (float)gi;
    fd[6]  = gys - (float)gj;
    fd[7]  = logf(sw / c_aw[bi]);
    fd[8]  = logf(sh / c_ah[bi]);
    fd[9]  = cls;
    fd[10] = 2.0f - wn * hn;
    flatArr[b * NT + t] = flat;
}

// ---------------- kernel 2: main loss ----------------
__global__ void __launch_bounds__(256)
yolo_loss_kernel(const float* __restrict__ x,
                 const float* __restrict__ fdata,
                 const int*   __restrict__ flatArr,
                 float* __restrict__ out,
                 float invB) {
    __shared__ float sm[NT][11];
    __shared__ int   sflat[NT];

    int b   = blockIdx.y;
    int tid = threadIdx.x;
    if (tid < NT * 11) sm[tid / 11][tid % 11] = fdata[(size_t)b * NT * 11 + tid];
    if (tid < NT)      sflat[tid] = flatArr[b * NT + tid];
    __syncthreads();

    int idx = blockIdx.x * blockDim.x + tid;

    float lLoc = 0.0f, lConf = 0.0f, lCls = 0.0f;

    if (idx < PP) {
        int a   = idx / HW;
        int rem = idx - a * HW;
        int i   = rem / WW;
        int j   = rem - i * WW;

        const float* base = x + ((size_t)b * (NA * (5 + NC)) + (size_t)a * (5 + NC)) * HW + rem;
        float z0 = base[0 * HW];
        float z1 = base[1 * HW];
        float z2 = base[2 * HW];
        float z3 = base[3 * HW];
        float z4 = base[4 * HW];

        float sx = sigmoidf(z0);
        float sy = sigmoidf(z1);

        // decoded box (stop-gradient path)
        float dx = (sx + (float)j) * STRIDE;
        float dy = (sy + (float)i) * STRIDE;
        float dw = expf(z2) * c_aw[a];
        float dh = expf(z3) * c_ah[a];
        float darea = dw * dh;
        float dx1 = dx - 0.5f * dw, dx2 = dx + 0.5f * dw;
        float dy1 = dy - 0.5f * dh, dy2 = dy + 0.5f * dh;

        // ignore mask: max IoU vs valid GT boxes
        float maxiou = 0.0f;
        #pragma unroll
        for (int t = 0; t < NT; ++t) {
            if (sm[t][0] > 0.5f) {
                float gx = sm[t][1], gy = sm[t][2], gw = sm[t][3], gh = sm[t][4];
                float lx = fmaxf(dx1, gx - 0.5f * gw);
                float rx = fminf(dx2, gx + 0.5f * gw);
                float ly = fmaxf(dy1, gy - 0.5f * gh);
                float ry = fminf(dy2, gy + 0.5f * gh);
                float iw = fmaxf(rx - lx, 0.0f);
                float ih = fmaxf(ry - ly, 0.0f);
                float inter = iw * ih;
                float iou = inter / (darea + gw * gh - inter + 1e-9f);
                maxiou = fmaxf(maxiou, iou);
            }
        }
        bool ignore = maxiou > 0.5f;

        // foreground: last target whose flat index hits this cell (scatter-set: last wins)
        int m = -1;
        #pragma unroll
        for (int t = 0; t < NT; ++t)
            if (sflat[t] == idx) m = t;
        bool fore = (m >= 0);

        // objectness BCE
        if (fore)        lConf = softplus_capped(-z4);      // target 1
        else if (!ignore) lConf = softplus_capped(z4);       // target 0 (background)

        if (fore) {
            // localization: 0.5 * w2 * [(sig(xy)-txy)^2 + (twh - t_wh)^2]
            float w2  = sm[m][10];
            float exy = (sx - sm[m][5]) * (sx - sm[m][5]) + (sy - sm[m][6]) * (sy - sm[m][6]);
            float ewh = (z2 - sm[m][7]) * (z2 - sm[m][7]) + (z3 - sm[m][8]) * (z3 - sm[m][8]);
            lLoc = 0.5f * w2 * (exy + ewh);

            // class BCE vs one-hot — only fore cells ever touch the 80 class channels
            int cid = (int)sm[m][9];
            float s = 0.0f;
            for (int c = 0; c < NC; ++c) {
                float z = base[(size_t)(5 + c) * HW];
                s += softplus_capped(c == cid ? -z : z);
            }
            lCls = s;
        }
    }

    lLoc *= invB; lConf *= invB; lCls *= invB;

    // wave32 reduction on the matrix pipe (EXEC is all-ones here: no early exits)
    float rLoc  = wave_reduce_wmma(lLoc);
    float rConf = wave_reduce_wmma(lConf);
    float rCls  = wave_reduce_wmma(lCls);

    if ((tid & 31) == 0) {
        atomicAdd(&out[0], rLoc);
        atomicAdd(&out[1], rConf);
        atomicAdd(&out[2], rCls);
    }
}

// ---------------- launcher ----------------
extern "C" void kernel_launch(void* const* d_in, const int* in_sizes, int n_in,
                              void* d_out, int out_size, void* d_ws, size_t ws_size,
                              hipStream_t stream) {
    const float* x   = (const float*)d_in[0];
    const float* tgt = (const float*)d_in[1];
    float* out = (float*)d_out;

    int B = in_sizes[0] / (NA * (5 + NC) * HW);   // 16 for the reference setup

    float* fdata  = (float*)d_ws;                         // B*NT*11 floats
    int*   flatA  = (int*)(fdata + (size_t)B * NT * 11);  // B*NT ints

    zero_out_kernel<<<1, 32, 0, stream>>>(out, out_size);
    match_kernel<<<B, 32, 0, stream>>>(tgt, fdata, flatA);

    dim3 grid((PP + 255) / 256, B);
    yolo_loss_kernel<<<grid, 256, 0, stream>>>(x, fdata, flatA, out, 1.0f / (float)B);
}